// NemoGPT_50268297232925
// MI455X (gfx1250) — compile-verified
//
#include <hip/hip_runtime.h>
#include <hip/hip_bf16.h>
#include <math.h>

typedef __attribute__((ext_vector_type(2))) float v2f;
typedef __attribute__((ext_vector_type(8))) float v8f;

constexpr int kE  = 64;     // embed dim
constexpr int kD  = 16;     // head dim
constexpr int kH  = 4;      // heads
constexpr int kL  = 3;      // layers
constexpr int kB  = 64;     // batch
constexpr int kT  = 64;     // seq
constexpr int kV  = 32000;  // vocab
constexpr int kFF = 256;    // 4*E
constexpr int kNTOK = kB * kT;
constexpr float kEPS = 1e-5f;
constexpr float kSCALE = 0.125f;  // 1/sqrt(E) = 1/8

// ---------------- WMMA f32 16x16x4 helpers (CDNA5 VGPR layouts, ISA 7.12.2) --

static __device__ __forceinline__ v8f wmma4(v2f a, v2f b, v8f c) {
  // D = A(16x4) * B(4x16) + C(16x16), full f32
  return __builtin_amdgcn_wmma_f32_16x16x4_f32(
      /*neg_a=*/false, a, /*neg_b=*/false, b,
      /*c_mod=*/(short)0, c, /*reuse_a=*/false, /*reuse_b=*/false);
}

// A fragment: 16x4 tile at (row0, k0) of row-major src with leading dim ld.
// lanes 0-15: M=lane, K={k0,k0+1}; lanes 16-31: M=lane-16, K={k0+2,k0+3}
static __device__ __forceinline__ v2f frag_a(const float* src, int ld, int row0, int k0) {
  int lane = threadIdx.x & 31;
  int r = row0 + (lane & 15);
  int k = k0 + ((lane >> 4) << 1);
  v2f a;
  a.x = src[r * ld + k];
  a.y = src[r * ld + k + 1];
  return a;
}

// B fragment: 4x16 tile at (k0, col0) of row-major src.
static __device__ __forceinline__ v2f frag_b(const float* src, int ld, int k0, int col0) {
  int lane = threadIdx.x & 31;
  int n = col0 + (lane & 15);
  int k = k0 + ((lane >> 4) << 1);
  v2f b;
  b.x = src[k * ld + n];
  b.y = src[(k + 1) * ld + n];
  return b;
}

// B fragment of a *transposed* operand: logical B[k][n] = srcT[n][k]
static __device__ __forceinline__ v2f frag_bt(const float* srcT, int ld, int k0, int col0) {
  int lane = threadIdx.x & 31;
  int n = col0 + (lane & 15);
  int k = k0 + ((lane >> 4) << 1);
  v2f b;
  b.x = srcT[n * ld + k];
  b.y = srcT[n * ld + k + 1];
  return b;
}

// Store 16x16 f32 C/D accumulator to row-major dst.
// VGPR r: lanes 0-15 -> M=r, lanes 16-31 -> M=r+8
static __device__ __forceinline__ void frag_store(float* dst, int ld, int row0, int col0, v8f c) {
  int lane = threadIdx.x & 31;
  int col = col0 + (lane & 15);
  int rbase = row0 + ((lane >> 4) << 3);
#pragma unroll
  for (int r = 0; r < 8; ++r) dst[(rbase + r) * ld + col] = c[r];
}

// ---------------- Kernel 1: token + position embedding ----------------------

__global__ void k_embed(const int* __restrict__ idx, const float* __restrict__ tok,
                        const float* __restrict__ pos, float* __restrict__ x) {
  int i = blockIdx.x * blockDim.x + threadIdx.x;   // over NTOK*E
  if (i >= kNTOK * kE) return;
  int e = i & (kE - 1);
  int t = i >> 6;            // flat token id 0..4095
  int tt = t & (kT - 1);     // position in sequence
  x[i] = tok[idx[t] * kE + e] + pos[tt * kE + e];
}

// ---------------- Kernel 2: fused attention block (per layer) ---------------
// One block per batch element b; 128 threads = 4 waves; wave w = head w.
// LDS: h[64][64] | h2[64][64] | per-wave q/k/v[64][16] | per-wave S[64][64]
// = 144 KB, fits the 320 KB WGP LDS.

__global__ void k_attn(float* __restrict__ x,
                       const float* __restrict__ Wq, const float* __restrict__ Wk,
                       const float* __restrict__ Wv, const float* __restrict__ Wp,
                       const float* __restrict__ bp, const float* __restrict__ g1,
                       const float* __restrict__ be1) {
  extern __shared__ float lds[];
  float* h  = lds;                 // 4096 floats
  float* h2 = h + kT * kE;         // 4096
  float* qb = h2 + kT * kE;        // 4 * 1024
  float* kb = qb + kH * kT * kD;
  float* vb = kb + kH * kT * kD;
  float* sb = vb + kH * kT * kD;   // 4 * 4096

  const int b = blockIdx.x;
  const int lane = threadIdx.x & 31;
  const int w = threadIdx.x >> 5;

  // LN1: one thread per token (threads 0..63)
  if (threadIdx.x < kT) {
    const float* xr = x + (b * kT + threadIdx.x) * kE;
    float mu = 0.f;
    for (int e = 0; e < kE; ++e) mu += xr[e];
    mu *= (1.0f / kE);
    float var = 0.f;
    for (int e = 0; e < kE; ++e) { float d = xr[e] - mu; var += d * d; }
    var *= (1.0f / kE);
    float inv = rsqrtf(var + kEPS);
    for (int e = 0; e < kE; ++e)
      h[threadIdx.x * kE + e] = (xr[e] - mu) * inv * g1[e] + be1[e];
  }
  __syncthreads();

  float* q = qb + w * kT * kD;
  float* kx = kb + w * kT * kD;
  float* v = vb + w * kT * kD;
  float* S = sb + w * kT * kT;
  const float* Wqh = Wq + w * kE * kD;
  const float* Wkh = Wk + w * kE * kD;
  const float* Wvh = Wv + w * kE * kD;

  // q/k/v = h @ W{q,k,v}[head]  (64x64 @ 64x16); A fragment shared 3x
  for (int mt = 0; mt < 4; ++mt) {
    v8f aq = {}; v8f ak = {}; v8f av = {};
    for (int ks = 0; ks < 16; ++ks) {
      v2f a = frag_a(h, kE, mt * 16, ks * 4);
      aq = wmma4(a, frag_b(Wqh, kD, ks * 4, 0), aq);
      ak = wmma4(a, frag_b(Wkh, kD, ks * 4, 0), ak);
      av = wmma4(a, frag_b(Wvh, kD, ks * 4, 0), av);
    }
    frag_store(q, kD, mt * 16, 0, aq);
    frag_store(kx, kD, mt * 16, 0, ak);
    frag_store(v, kD, mt * 16, 0, av);
  }
  __syncthreads();

  // S = q @ k^T  (64x16 @ 16x64); transpose via frag_bt indexing
  for (int mt = 0; mt < 4; ++mt)
    for (int nt = 0; nt < 4; ++nt) {
      v8f acc = {};
      for (int ks = 0; ks < 4; ++ks)
        acc = wmma4(frag_a(q, kD, mt * 16, ks * 4),
                    frag_bt(kx, kD, ks * 4, nt * 16), acc);
      frag_store(S, kT, mt * 16, nt * 16, acc);
    }
  __syncthreads();

  // causal softmax (scale 1/8): 2 rows per lane
  for (int r = lane; r < kT; r += 32) {
    float m = -1e30f;
    for (int j = 0; j <= r; ++j) m = fmaxf(m, S[r * kT + j] * kSCALE);
    float sum = 0.f;
    for (int j = 0; j <= r; ++j) {
      float e2 = __expf(S[r * kT + j] * kSCALE - m);
      S[r * kT + j] = e2;
      sum += e2;
    }
    float inv = 1.0f / sum;
    for (int j = 0; j <= r; ++j) S[r * kT + j] *= inv;
    for (int j = r + 1; j < kT; ++j) S[r * kT + j] = 0.0f;
  }
  __syncthreads();

  // attn = S @ v -> h2[:, w*16 .. w*16+16)
  for (int mt = 0; mt < 4; ++mt) {
    v8f acc = {};
    for (int ks = 0; ks < 16; ++ks)
      acc = wmma4(frag_a(S, kT, mt * 16, ks * 4),
                  frag_b(v, kD, ks * 4, 0), acc);
    frag_store(h2, kE, mt * 16, w * kD, acc);
  }
  __syncthreads();

  // out = h2 @ Wproj + bproj, residual into x; wave w owns cols [w*16, w*16+16)
  for (int mt = 0; mt < 4; ++mt) {
    v8f acc = {};
    for (int ks = 0; ks < 16; ++ks)
      acc = wmma4(frag_a(h2, kE, mt * 16, ks * 4),
                  frag_b(Wp, kE, ks * 4, w * 16), acc);
    int col = w * 16 + (lane & 15);
    int rbase = mt * 16 + ((lane >> 4) << 3);
    float bias = bp[col];
#pragma unroll
    for (int r = 0; r < 8; ++r)
      x[(b * kT + rbase + r) * kE + col] += acc[r] + bias;
  }
}

// ---------------- Kernel 3: fused MLP block (per layer) ---------------------
// One block per batch element; 128 threads = 4 waves.
// LDS: h[64][64] | g[64][256] (post-GELU hidden) = 80 KB.

__global__ void k_mlp(float* __restrict__ x,
                      const float* __restrict__ W1, const float* __restrict__ bb1,
                      const float* __restrict__ W2, const float* __restrict__ bb2,
                      const float* __restrict__ g2, const float* __restrict__ be2) {
  extern __shared__ float lds[];
  float* h = lds;              // 4096
  float* g = h + kT * kE;      // 16384

  const int b = blockIdx.x;
  const int lane = threadIdx.x & 31;
  const int w = threadIdx.x >> 5;

  // LN2
  if (threadIdx.x < kT) {
    const float* xr = x + (b * kT + threadIdx.x) * kE;
    float mu = 0.f;
    for (int e = 0; e < kE; ++e) mu += xr[e];
    mu *= (1.0f / kE);
    float var = 0.f;
    for (int e = 0; e < kE; ++e) { float d = xr[e] - mu; var += d * d; }
    var *= (1.0f / kE);
    float inv = rsqrtf(var + kEPS);
    for (int e = 0; e < kE; ++e)
      h[threadIdx.x * kE + e] = (xr[e] - mu) * inv * g2[e] + be2[e];
  }
  __syncthreads();

  // g = gelu(h @ W1 + b1); wave w owns hidden cols [w*64, w*64+64)
  for (int nt = 0; nt < 4; ++nt) {
    int col0 = w * 64 + nt * 16;
    for (int mt = 0; mt < 4; ++mt) {
      v8f acc = {};
      for (int ks = 0; ks < 16; ++ks)
        acc = wmma4(frag_a(h, kE, mt * 16, ks * 4),
                    frag_b(W1, kFF, ks * 4, col0), acc);
      int col = col0 + (lane & 15);
      int rbase = mt * 16 + ((lane >> 4) << 3);
      float bias = bb1[col];
#pragma unroll
      for (int r = 0; r < 8; ++r) {
        float t = acc[r] + bias;                       // exact GELU (erf form)
        g[(rbase + r) * kFF + col] = 0.5f * t * (1.0f + erff(t * 0.70710678118f));
      }
    }
  }
  __syncthreads();

  // out = g @ W2 + b2, residual; wave w owns cols [w*16, w*16+16)
  for (int mt = 0; mt < 4; ++mt) {
    v8f acc = {};
    for (int ks = 0; ks < 64; ++ks)
      acc = wmma4(frag_a(g, kFF, mt * 16, ks * 4),
                  frag_b(W2, kE, ks * 4, w * 16), acc);
    int col = w * 16 + (lane & 15);
    int rbase = mt * 16 + ((lane >> 4) << 3);
    float bias = bb2[col];
#pragma unroll
    for (int r = 0; r < 8; ++r)
      x[(b * kT + rbase + r) * kE + col] += acc[r] + bias;
  }
}

// ---------------- Kernel 4: final LayerNorm ---------------------------------

__global__ void k_lnf(const float* __restrict__ x, const float* __restrict__ g,
                      const float* __restrict__ be, float* __restrict__ xln) {
  int t = blockIdx.x * blockDim.x + threadIdx.x;
  if (t >= kNTOK) return;
  const float* xr = x + t * kE;
  float mu = 0.f;
  for (int e = 0; e < kE; ++e) mu += xr[e];
  mu *= (1.0f / kE);
  float var = 0.f;
  for (int e = 0; e < kE; ++e) { float d = xr[e] - mu; var += d * d; }
  var *= (1.0f / kE);
  float inv = rsqrtf(var + kEPS);
  for (int e = 0; e < kE; ++e)
    xln[t * kE + e] = (xr[e] - mu) * inv * g[e] + be[e];
}

// ---------------- Kernel 5: LM head (store-bound, 524 MB out) ---------------
// Block: 256 threads = 8 waves; block covers 16 rows x 512 cols.
// The block's 16x64 A tile of xln is staged once into LDS (4 KB), so waves
// pull A fragments from LDS while VMEM streams Wlm (L2-resident, 8.2 MB)
// and writes logits. Logits are written NON-TEMPORAL (TH=NT): the 524 MB
// output is never re-read, so it must not displace Wlm in the 192 MB L2.

__global__ void k_lmhead(const float* __restrict__ xln, const float* __restrict__ Wlm,
                         const float* __restrict__ blm, float* __restrict__ out) {
  __shared__ float atile[16 * kE];   // 16 rows x 64 cols of xln

  const int lane = threadIdx.x & 31;
  const int w = threadIdx.x >> 5;
  const int row0 = blockIdx.y * 16;
  const int colbase = blockIdx.x * 512 + w * 64;

  // cooperative stage: 1024 floats / 256 threads = 4 each (coalesced b128-able)
  {
    const float* src = xln + (size_t)row0 * kE;
#pragma unroll
    for (int i = threadIdx.x; i < 16 * kE; i += 256) atile[i] = src[i];
  }
  __syncthreads();

  if (colbase < kV) {                // tail block: waves past V skip compute
    v8f acc[4] = {};
    for (int ks = 0; ks < 16; ++ks) {
      v2f a = frag_a(atile, kE, 0, ks * 4);    // A from LDS
#pragma unroll
      for (int nt = 0; nt < 4; ++nt)
        acc[nt] = wmma4(a, frag_b(Wlm, kV, ks * 4, colbase + nt * 16), acc[nt]);
    }
#pragma unroll
    for (int nt = 0; nt < 4; ++nt) {
      int col = colbase + nt * 16 + (lane & 15);
      int rbase = row0 + ((lane >> 4) << 3);
      float bias = blm[col];
#pragma unroll
      for (int r = 0; r < 8; ++r)
        __builtin_nontemporal_store(acc[nt][r] + bias,
                                    out + (size_t)(rbase + r) * kV + col);
    }
  }
}

// ---------------- Launch ----------------------------------------------------

extern "C" void kernel_launch(void* const* d_in, const int* in_sizes, int n_in,
                              void* d_out, int out_size, void* d_ws, size_t ws_size,
                              hipStream_t stream) {
  const int*   idx  = (const int*)  d_in[0];
  const float* tok  = (const float*)d_in[1];
  const float* pos  = (const float*)d_in[2];
  const float* Wq   = (const float*)d_in[3];
  const float* Wk   = (const float*)d_in[4];
  const float* Wv   = (const float*)d_in[5];
  const float* Wp   = (const float*)d_in[6];
  const float* bp   = (const float*)d_in[7];
  const float* ln1g = (const float*)d_in[8];
  const float* ln1b = (const float*)d_in[9];
  const float* ln2g = (const float*)d_in[10];
  const float* ln2b = (const float*)d_in[11];
  const float* W1   = (const float*)d_in[12];
  const float* b1   = (const float*)d_in[13];
  const float* W2   = (const float*)d_in[14];
  const float* b2   = (const float*)d_in[15];
  const float* lnfg = (const float*)d_in[16];
  const float* lnfb = (const float*)d_in[17];
  const float* Wlm  = (const float*)d_in[18];
  const float* blm  = (const float*)d_in[19];
  float* out = (float*)d_out;

  float* xbuf = (float*)d_ws;          // [4096, 64] residual stream
  float* xln  = xbuf + kNTOK * kE;     // [4096, 64] final LN output

  k_embed<<<(kNTOK * kE + 255) / 256, 256, 0, stream>>>(idx, tok, pos, xbuf);

  constexpr size_t ldsAttn = (size_t)(2 * kT * kE + 3 * kH * kT * kD + kH * kT * kT) * 4; // 147456 B
  constexpr size_t ldsMlp  = (size_t)(kT * kE + kT * kFF) * 4;                            // 81920 B

  for (int l = 0; l < kL; ++l) {
    k_attn<<<kB, 128, ldsAttn, stream>>>(xbuf,
        Wq + (size_t)l * kH * kE * kD, Wk + (size_t)l * kH * kE * kD,
        Wv + (size_t)l * kH * kE * kD, Wp + (size_t)l * kE * kE,
        bp + (size_t)l * kE, ln1g + (size_t)l * kE, ln1b + (size_t)l * kE);
    k_mlp<<<kB, 128, ldsMlp, stream>>>(xbuf,
        W1 + (size_t)l * kE * kFF, b1 + (size_t)l * kFF,
        W2 + (size_t)l * kFF * kE, b2 + (size_t)l * kE,
        ln2g + (size_t)l * kE, ln2b + (size_t)l * kE);
  }

  k_lnf<<<kNTOK / 64, 64, 0, stream>>>(xbuf, lnfg, lnfb, xln);

  // 63 x 512 cols >= 32000; 256 row tiles of 16
  k_lmhead<<<dim3(63, 256), 256, 0, stream>>>(xln, Wlm, blm, out);
}